// EntmaxAlpha_29497835389384
// MI455X (gfx1250) — compile-verified
//
#include <hip/hip_runtime.h>
#include <math.h>

#ifndef __has_builtin
#define __has_builtin(x) 0
#endif

#define KDIM   1024          // last-dim length (reference: K = 1024)
#define WAVES  8             // waves per block; one row per wave
#define EPL    (KDIM / 32)   // elements per lane = 32
#define NIT    50            // bisection iterations (reference N_ITER)

typedef unsigned int u32x4 __attribute__((ext_vector_type(4)));
typedef int          i32x4 __attribute__((ext_vector_type(4)));
typedef int          i32x8 __attribute__((ext_vector_type(8)));

#if defined(__HIP_DEVICE_COMPILE__) && defined(__AMDGCN__) &&            \
    __has_builtin(__builtin_amdgcn_tensor_load_to_lds) &&                \
    __has_builtin(__builtin_amdgcn_s_wait_tensorcnt)
#define HAVE_TDM 1
#else
#define HAVE_TDM 0
#endif

// v_log_f32 / v_exp_f32 are base-2 on AMDGCN: t^p = exp2(p * log2(t)), t >= 0.
// log2(0) = -inf -> exp2(-inf) = 0, which is exactly the needed 0^p behavior.
__device__ __forceinline__ float fast_log2(float x) {
#if defined(__HIP_DEVICE_COMPILE__) && __has_builtin(__builtin_amdgcn_logf)
  return __builtin_amdgcn_logf(x);     // v_log_f32 (base-2)
#else
  return log2f(x);                     // host pass / fallback
#endif
}
__device__ __forceinline__ float fast_exp2(float x) {
#if defined(__HIP_DEVICE_COMPILE__) && __has_builtin(__builtin_amdgcn_exp2f)
  return __builtin_amdgcn_exp2f(x);    // v_exp_f32 (base-2)
#else
  return exp2f(x);                     // host pass / fallback
#endif
}
__device__ __forceinline__ float pow_pos(float t, float pexp) {
  return fast_exp2(pexp * fast_log2(t));
}

// wave32 butterfly reductions (all lanes end with the result)
__device__ __forceinline__ float wave_sum32(float v) {
#pragma unroll
  for (int m = 16; m >= 1; m >>= 1) v += __shfl_xor(v, m, 32);
  return v;
}
__device__ __forceinline__ float wave_max32(float v) {
#pragma unroll
  for (int m = 16; m >= 1; m >>= 1) v = fmaxf(v, __shfl_xor(v, m, 32));
  return v;
}

#if HAVE_TDM
// Issue a Tensor Data Mover DMA: one contiguous row of KDIM fp32 from global
// memory into this wave's LDS slice. D# layout per CDNA5 ISA ch.8 (§8.3/§8.4):
//   group0: count=1 | lds_addr | global_addr[56:0] | type=2
//   group1: data_size=4B, tensor_dim0=K, tensor_dim1=1, tile=(K,1),
//           tensor_dim0_stride=K; no multicast / padding / iteration.
__device__ __forceinline__ void tdm_load_row(const float* gsrc, unsigned lds_byte) {
  unsigned long long ga = (unsigned long long)gsrc;
  u32x4 g0;
  g0[0] = 1u;                                                // count=1, user D#
  g0[1] = lds_byte;                                          // lds_addr (bytes)
  g0[2] = (unsigned)ga;                                      // global_addr[31:0]
  g0[3] = ((unsigned)(ga >> 32) & 0x01FFFFFFu) | (2u << 30); // addr[56:32]|type=2
  i32x8 g1;
  g1[0] = 0x00020000;                                        // data_size = 2 (4 bytes)
  g1[1] = (int)(KDIM << 16);                                 // tensor_dim0[15:0]
  g1[2] = (int)(1 << 16);                                    // dim0 hi=0, tensor_dim1=1
  g1[3] = (int)(KDIM << 16);                                 // tile_dim0 = K
  g1[4] = 1;                                                 // tile_dim1=1, tile_dim2=0
  g1[5] = KDIM;                                              // tensor_dim0_stride lo
  g1[6] = 0;
  g1[7] = 0;
  i32x4 gz = {0, 0, 0, 0};
#if __clang_major__ >= 23
  i32x8 gz8 = {0, 0, 0, 0, 0, 0, 0, 0};
  __builtin_amdgcn_tensor_load_to_lds(g0, g1, gz, gz, gz8, 0);
#else
  __builtin_amdgcn_tensor_load_to_lds(g0, g1, gz, gz, 0);
#endif
}
#endif

__global__ __launch_bounds__(WAVES * 32) void entmax_bisect_rows_kernel(
    const float* __restrict__ x, const float* __restrict__ alpha,
    float* __restrict__ out, int nrows, int H) {
  __shared__ __align__(16) float sh[WAVES * KDIM];

  const int wave = (int)(threadIdx.x >> 5);
  const int lane = (int)(threadIdx.x & 31);
  const int row  = (int)blockIdx.x * WAVES + wave;   // wave-uniform
  if (row >= nrows) return;

  float* shrow = &sh[wave * KDIM];
  const float* src = x + (size_t)row * KDIM;

#if HAVE_TDM
  // Per-wave async DMA into this wave's private LDS slice: no barrier needed,
  // each wave starts compute as soon as its own row lands.
  tdm_load_row(src, (unsigned)(unsigned long long)shrow);
  __builtin_amdgcn_s_wait_tensorcnt(0);
  asm volatile("" ::: "memory");
#else
  // Fallback: coalesced float4 copy (ds ordering handled by same-wave DScnt).
#pragma unroll
  for (int j = 0; j < EPL / 4; ++j)
    ((float4*)shrow)[lane + 32 * j] = ((const float4*)src)[lane + 32 * j];
  asm volatile("" ::: "memory");
#endif

  // per-head alpha: rows per (b,h) = Q = 1024  ->  h = (row / 1024) % H
  const int h = (row >> 10) % H;
  const float am1  = alpha[h] - 1.0f;
  const float pexp = 1.0f / am1;

  // LDS -> registers, scale by (alpha-1), track lane-local max.
  float xs[EPL];
  float mx = -3.402823466e38f;
#pragma unroll
  for (int j = 0; j < EPL / 4; ++j) {
    float4 v = ((const float4*)shrow)[lane + 32 * j];
    float a0 = v.x * am1, a1 = v.y * am1, a2 = v.z * am1, a3 = v.w * am1;
    xs[4 * j + 0] = a0; xs[4 * j + 1] = a1;
    xs[4 * j + 2] = a2; xs[4 * j + 3] = a3;
    mx = fmaxf(mx, fmaxf(fmaxf(a0, a1), fmaxf(a2, a3)));
  }
  mx = wave_max32(mx);

  float tau_lo = mx - 1.0f;
  // (1/d)^(alpha-1) with d=1024  ==  exp2(-10*(alpha-1))  (no pow needed)
  const float tau_hi = mx - fast_exp2(-10.0f * am1);

  float s = 0.0f;
#pragma unroll
  for (int i = 0; i < EPL; ++i)
    s += pow_pos(fmaxf(xs[i] - tau_lo, 0.0f), pexp);
  const float f_lo = wave_sum32(s) - 1.0f;

  float dm    = tau_hi - tau_lo;
  float tau_m = tau_lo;
  for (int it = 0; it < NIT; ++it) {
    dm *= 0.5f;
    tau_m = tau_lo + dm;
    float sm = 0.0f;
#pragma unroll
    for (int i = 0; i < EPL; ++i)
      sm += pow_pos(fmaxf(xs[i] - tau_m, 0.0f), pexp);
    const float f_m = wave_sum32(sm) - 1.0f;          // wave-uniform
    tau_lo = (f_m * f_lo >= 0.0f) ? tau_m : tau_lo;   // uniform select
  }

  // Reference returns p_m from the LAST iteration's midpoint, normalized.
  float pv[EPL];
  float tot = 0.0f;
#pragma unroll
  for (int i = 0; i < EPL; ++i) {
    pv[i] = pow_pos(fmaxf(xs[i] - tau_m, 0.0f), pexp);
    tot += pv[i];
  }
  tot = wave_sum32(tot);
  const float inv = 1.0f / tot;

  float4* op = (float4*)(out + (size_t)row * KDIM);
#pragma unroll
  for (int j = 0; j < EPL / 4; ++j) {
    float4 v;
    v.x = pv[4 * j + 0] * inv;
    v.y = pv[4 * j + 1] * inv;
    v.z = pv[4 * j + 2] * inv;
    v.w = pv[4 * j + 3] * inv;
    op[lane + 32 * j] = v;
  }
}

extern "C" void kernel_launch(void* const* d_in, const int* in_sizes, int n_in,
                              void* d_out, int out_size, void* d_ws, size_t ws_size,
                              hipStream_t stream) {
  (void)n_in; (void)out_size; (void)d_ws; (void)ws_size;
  const float* x     = (const float*)d_in[0];   // att_scores [B,H,Q,K] fp32
  const float* alpha = (const float*)d_in[1];   // alpha [H] fp32
  float* out = (float*)d_out;

  const int H     = in_sizes[1];
  const int nrows = in_sizes[0] / KDIM;         // B*H*Q
  const int blocks = (nrows + WAVES - 1) / WAVES;

  entmax_bisect_rows_kernel<<<blocks, WAVES * 32, 0, stream>>>(x, alpha, out, nrows, H);
}